// CrossAttentionFeatureBlock_53893249630437
// MI455X (gfx1250) — compile-verified
//
#include <hip/hip_runtime.h>
#include <math.h>

typedef __attribute__((ext_vector_type(2))) float v2f;
typedef __attribute__((ext_vector_type(8))) float v8f;

#define BATCH 32
#define LA 2048
#define LB 2048
#define TPB 256
#define WAVES 8
#define ROWS_PER_BLOCK 128            // 8 waves x 16 rows
#define TILES_PER_BATCH (LA / ROWS_PER_BLOCK)

__global__ __launch_bounds__(TPB) void cross_attn_wmma_kernel(
    const float* __restrict__ a, const float* __restrict__ b,
    const float* __restrict__ ipw, const float* __restrict__ ipb,
    const float* __restrict__ ow,  const float* __restrict__ ob,
    float* __restrict__ out)
{
    __shared__ __align__(16) float s_k[LB];   // projected keys
    __shared__ __align__(16) float s_v[LB];   // projected values
    __shared__ float s_red[2 * TPB];          // max / min reduction
    __shared__ float s_c[WAVES][32][8];       // accumulator spill per wave

    const int tid   = threadIdx.x;
    const int wave  = tid >> 5;
    const int lane  = tid & 31;
    const int batch = blockIdx.x / TILES_PER_BATCH;
    const int tile  = blockIdx.x % TILES_PER_BATCH;

    const float wq = ipw[0], wk = ipw[1], wv = ipw[2];
    const float bq = ipb[0], bk = ipb[1], bv = ipb[2];
    const float w_out = ow[0], b_out = ob[0];

    // ---- Stage k,v into LDS; track per-batch kmax/kmin for stable softmax ----
    float lmax = -INFINITY, lmin = INFINITY;
    const float* brow = b + (size_t)batch * LB;
    for (int i = tid; i < LB; i += TPB) {
        float bi = brow[i];
        float kv = fmaf(bi, wk, bk);
        s_k[i] = kv;
        s_v[i] = fmaf(bi, wv, bv);
        lmax = fmaxf(lmax, kv);
        lmin = fminf(lmin, kv);
    }
    s_red[tid]       = lmax;
    s_red[TPB + tid] = lmin;
    __syncthreads();
    for (int s = TPB / 2; s > 0; s >>= 1) {
        if (tid < s) {
            s_red[tid]       = fmaxf(s_red[tid],       s_red[tid + s]);
            s_red[TPB + tid] = fminf(s_red[TPB + tid], s_red[TPB + tid + s]);
        }
        __syncthreads();
    }
    const float kmax = s_red[0], kmin = s_red[TPB];

    // ---- Per-lane row state (16-row tile per wave) ----
    const int   rowBase = tile * ROWS_PER_BLOCK + wave * 16;
    const int   myRow   = rowBase + (lane & 15);
    const float qa = a[(size_t)batch * LA + myRow];
    const float q  = fmaf(qa, wq, bq);
    // row max of q*k_j is q*kmax (q>=0) or q*kmin (q<0)
    const float m  = (q >= 0.0f) ? q * kmax : q * kmin;
    const float L2E = 1.4426950408889634f;
    const float ql2  = q * L2E;
    const float nml2 = -(m * L2E);

    // A layout (16x4 f32): lanes 0-15 -> K=0,1 ; lanes 16-31 -> K=2,3
    const int  lsel = lane >> 4;              // 0 or 1 -> K pair select
    const int  nsel = lane & 15;              // N column this lane supplies in B
    const bool isn0   = (nsel == 0);
    const float bcnst = (nsel == 1) ? 1.0f : 0.0f;

    const float2* s_k2 = reinterpret_cast<const float2*>(s_k);
    const float2* s_v2 = reinterpret_cast<const float2*>(s_v);

    v8f c = {0.f, 0.f, 0.f, 0.f, 0.f, 0.f, 0.f, 0.f};

    // ---- Main loop: 4 keys per WMMA step; D[:,0]=Σ e*v, D[:,1]=Σ e ----
    #pragma unroll 8
    for (int jb = 0; jb < LB; jb += 4) {
        int idx = (jb >> 1) + lsel;           // float2 index: ds_load_b64
        float2 kk = s_k2[idx];
        float2 vv = s_v2[idx];
        // raw v_exp_f32: args <= 0; sub-(-126) terms are zero in the softmax sum
        float e0 = __builtin_amdgcn_exp2f(fmaf(ql2, kk.x, nml2));
        float e1 = __builtin_amdgcn_exp2f(fmaf(ql2, kk.y, nml2));
        v2f A; A.x = e0; A.y = e1;
        v2f B;
        B.x = isn0 ? vv.x : bcnst;
        B.y = isn0 ? vv.y : bcnst;
        c = __builtin_amdgcn_wmma_f32_16x16x4_f32(
                /*neg_a=*/false, A, /*neg_b=*/false, B,
                /*c_mod=*/(short)0, c, /*reuse_a=*/false, /*reuse_b=*/false);
    }

    // ---- Extract: row m numer at (VGPR m%8, lane 0/16), denom at lane 1/17 ----
    #pragma unroll
    for (int g = 0; g < 8; ++g) s_c[wave][lane][g] = c[g];
    // same-wave LDS RAW: in-order, compiler inserts s_wait_dscnt

    if (lane < 16) {
        int srcLane = (lane < 8) ? 0 : 16;
        int g = lane & 7;
        float numer = s_c[wave][srcLane][g];
        float denom = s_c[wave][srcLane + 1][g];
        float ctx = numer / denom;
        float x = fmaf(ctx, w_out, b_out);
        float y = x + qa;                          // residual (qa == a[myRow] here)
        float r = 0.5f * y * (1.0f + erff(y * 0.70710678118654752f)); // exact GELU
        out[(size_t)batch * LA + rowBase + lane] = r;
    }
}

extern "C" void kernel_launch(void* const* d_in, const int* in_sizes, int n_in,
                              void* d_out, int out_size, void* d_ws, size_t ws_size,
                              hipStream_t stream) {
    const float* a   = (const float*)d_in[0];
    const float* b   = (const float*)d_in[1];
    const float* ipw = (const float*)d_in[2];
    const float* ipb = (const float*)d_in[3];
    const float* ow  = (const float*)d_in[4];
    const float* ob  = (const float*)d_in[5];
    float* out = (float*)d_out;
    (void)in_sizes; (void)n_in; (void)out_size; (void)d_ws; (void)ws_size;

    dim3 grid(BATCH * TILES_PER_BATCH);   // 512 blocks
    dim3 block(TPB);                      // 8 wave32 per block
    hipLaunchKernelGGL(cross_attn_wmma_kernel, grid, block, 0, stream,
                       a, b, ipw, ipb, ow, ob, out);
}